// PGE_62766652064245
// MI455X (gfx1250) — compile-verified
//
#include <hip/hip_runtime.h>

typedef __attribute__((ext_vector_type(2))) float v2f;
typedef __attribute__((ext_vector_type(8))) float v8f;

#define N_S 384           // samples
#define D_F 256           // feature dim (K)
#define Q_P 256           // pivot queue
#define C_C 345           // classes
#define M_D 3             // domains
#define QCM (Q_P * C_C * M_D)
#define MAXT 26           // max 16-row tiles after per-domain padding (24 + 2)

// ---------------------------------------------------------------------------
// One wave (32 lanes) per 256-float row: out[row] = sum(row^2).
// Used for both emb (rowsPerSrc=N) and pivot (rowsPerSrc=Q*C*M) norms.
// blockDim = 256 (8 waves/block).
// ---------------------------------------------------------------------------
__global__ void rowsq_kernel(const float* __restrict__ a0,
                             const float* __restrict__ a1,
                             float* __restrict__ out, int rowsPerSrc) {
  int wid  = (int)((blockIdx.x * blockDim.x + threadIdx.x) >> 5);
  int lane = threadIdx.x & 31;
  int total = rowsPerSrc * 2;
  if (wid >= total) return;
  int p = wid / rowsPerSrc;
  size_t row = (size_t)(wid % rowsPerSrc);
  const float* src = (p == 0 ? a0 : a1) + row * D_F;
  float s = 0.f;
#pragma unroll
  for (int i = 0; i < 2; ++i) {
    float4 v = ((const float4*)src)[lane + 32 * i];
    s += v.x * v.x + v.y * v.y + v.z * v.z + v.w * v.w;
  }
#pragma unroll
  for (int off = 16; off >= 1; off >>= 1) s += __shfl_xor(s, off, 32);
  if (lane == 0) out[(size_t)p * rowsPerSrc + row] = s;
}

// ---------------------------------------------------------------------------
// Deterministic counting sort of samples by domain + 16-row tile descriptors.
// desc[t] = {m, startInPerm, validRows}. meta[0] = #tiles.
// ---------------------------------------------------------------------------
__global__ void partition_kernel(const int* __restrict__ all_d,
                                 int* __restrict__ perm,
                                 int* __restrict__ desc,
                                 int* __restrict__ meta) {
  if (threadIdx.x != 0 || blockIdx.x != 0) return;
  int cnt[M_D];
  for (int m = 0; m < M_D; ++m) cnt[m] = 0;
  for (int n = 0; n < N_S; ++n) cnt[all_d[n]]++;
  int off[M_D], pos[M_D];
  off[0] = 0;
  for (int m = 1; m < M_D; ++m) off[m] = off[m - 1] + cnt[m - 1];
  for (int m = 0; m < M_D; ++m) pos[m] = off[m];
  for (int n = 0; n < N_S; ++n) perm[pos[all_d[n]]++] = n;  // stable
  int nt = 0;
  for (int m = 0; m < M_D; ++m) {
    for (int t = 0; t * 16 < cnt[m]; ++t) {
      int rem = cnt[m] - t * 16;
      desc[nt * 3 + 0] = m;
      desc[nt * 3 + 1] = off[m] + t * 16;
      desc[nt * 3 + 2] = rem < 16 ? rem : 16;
      ++nt;
    }
  }
  meta[0] = nt;
}

// ---------------------------------------------------------------------------
// Main fused GEMM+epilogue kernel, one wave32 per block.
// block = (tile, c, pair). Tiles are domain-homogeneous (m from descriptor),
// so only the needed [N, C, Q] slice of the distance tensor is computed
// (3x less matrix work than the dense [N, M, C, Q]). 64 chained
// v_wmma_f32_16x16x4_f32 per 16-wide q tile; min over q in registers;
// rep-term mask (y != c) and one fp64 atomic per block.
// ---------------------------------------------------------------------------
__global__ __launch_bounds__(32)
void dist_wmma_kernel(const float* __restrict__ emb0, const float* __restrict__ emb1,
                      const float* __restrict__ piv0, const float* __restrict__ piv1,
                      const float* __restrict__ en2,  const float* __restrict__ pn2,
                      const int* __restrict__ all_y,
                      const int* __restrict__ perm,
                      const int* __restrict__ desc,
                      const int* __restrict__ meta,
                      double* __restrict__ acc) {
  const int nt = meta[0];
  const int t  = blockIdx.x;
  if (t >= nt) return;                  // uniform scalar branch, pre-barrier

  const int lane = threadIdx.x;         // 0..31
  const int lq   = lane & 15;
  const int half = lane >> 4;
  const int c    = blockIdx.y;
  const int p    = blockIdx.z;

  const int m     = desc[t * 3 + 0];
  const int start = desc[t * 3 + 1];
  const int cnt   = desc[t * 3 + 2];

  const float* emb  = p ? emb1 : emb0;
  const float* piv  = p ? piv1 : piv0;
  const float* en2p = en2 + (size_t)p * N_S;
  const float* pn2p = pn2 + (size_t)p * QCM;

  __shared__ float tileA[16 * D_F];
  __shared__ int   sRow[16];
  __shared__ float e2Row[16];

  if (lane < 16) {
    int s = (lane < cnt) ? perm[start + lane] : -1;
    sRow[lane]  = s;
    e2Row[lane] = (s >= 0) ? en2p[s] : 0.f;
  }
  __syncthreads();

  // Gather the 16 sample rows into LDS (padding rows zero-filled).
  for (int i = 0; i < 16; ++i) {
    int s = sRow[i];
#pragma unroll
    for (int j = 0; j < 2; ++j) {
      float4 v = make_float4(0.f, 0.f, 0.f, 0.f);
      if (s >= 0) v = ((const float4*)(emb + (size_t)s * D_F))[lane + 32 * j];
      ((float4*)(tileA + i * D_F))[lane + 32 * j] = v;
    }
  }
  __syncthreads();

  float en2v[8];
#pragma unroll
  for (int r = 0; r < 8; ++r) en2v[r] = e2Row[r + 8 * half];

  float minv[8];
#pragma unroll
  for (int r = 0; r < 8; ++r) minv[r] = 3.4e38f;

  // A fragment: lane l holds (M = l%16, K = 2*half + {0,1})
  const float* aBase = tileA + (size_t)lq * D_F + 2 * half;

  for (int qt = 0; qt < 16; ++qt) {
    const int q = qt * 16 + lq;
    // B fragment: lane l holds (N column = l%16, K = 2*half + {0,1})
    const float* bBase =
        piv + (((size_t)q * C_C + c) * M_D + m) * D_F + 2 * half;

    v8f accf = {0.f, 0.f, 0.f, 0.f, 0.f, 0.f, 0.f, 0.f};
#pragma unroll 8
    for (int kk = 0; kk < 64; ++kk) {
      v2f a = *(const v2f*)(aBase + 4 * kk);     // LDS, 8B aligned
      v2f b = *(const v2f*)(bBase + 4 * kk);     // global, 8B aligned
      accf = __builtin_amdgcn_wmma_f32_16x16x4_f32(
          /*neg_a=*/false, a, /*neg_b=*/false, b,
          /*c_mod=*/(short)0, accf, /*reuse_a=*/false, /*reuse_b=*/false);
    }

    const float p2 = pn2p[((size_t)q * C_C + c) * M_D + m];
#pragma unroll
    for (int r = 0; r < 8; ++r) {
      float d2 = en2v[r] + p2 - 2.0f * accf[r];
      float dd = sqrtf(fmaxf(d2, 0.0f));
      minv[r] = fminf(minv[r], dd);
    }
  }

  // Min over the 16 q-columns striped across each half-wave's lanes.
#pragma unroll
  for (int r = 0; r < 8; ++r) {
#pragma unroll
    for (int off = 1; off < 16; off <<= 1)
      minv[r] = fminf(minv[r], __shfl_xor(minv[r], off, 32));
  }

  float repPart = 0.f;
  if (lq == 0) {  // lanes 0 and 16: 8 sample rows each
#pragma unroll
    for (int r = 0; r < 8; ++r) {
      int s = sRow[r + 8 * half];
      if (s >= 0 && all_y[s] != c) repPart += minv[r];
    }
  }
  repPart += __shfl_xor(repPart, 16, 32);
  if (lane == 0 && repPart != 0.f) atomicAdd(&acc[0], (double)repPart);
}

// ---------------------------------------------------------------------------
// att term: one wave per (sample, domain, pair); c = all_y[n].
// Only 0.6% of total FLOPs -> plain VALU float4 dot products.
// ---------------------------------------------------------------------------
__global__ __launch_bounds__(32)
void att_kernel(const float* __restrict__ emb0, const float* __restrict__ emb1,
                const float* __restrict__ piv0, const float* __restrict__ piv1,
                const float* __restrict__ en2,  const float* __restrict__ pn2,
                const int* __restrict__ all_y, double* __restrict__ acc) {
  const int s    = blockIdx.x;
  const int m    = blockIdx.y;
  const int p    = blockIdx.z;
  const int lane = threadIdx.x;

  const float* emb  = p ? emb1 : emb0;
  const float* piv  = p ? piv1 : piv0;
  const float* pn2p = pn2 + (size_t)p * QCM;
  const int   c  = all_y[s];
  const float e2 = en2[p * N_S + s];

  __shared__ float erow[D_F];
#pragma unroll
  for (int j = 0; j < 2; ++j)
    ((float4*)erow)[lane + 32 * j] =
        ((const float4*)(emb + (size_t)s * D_F))[lane + 32 * j];
  __syncthreads();

  float mx = -1.f;
  for (int j = 0; j < 8; ++j) {
    const int q = lane + 32 * j;
    const float4* prow =
        (const float4*)(piv + (((size_t)q * C_C + c) * M_D + m) * D_F);
    float dot = 0.f;
#pragma unroll 16
    for (int k = 0; k < 64; ++k) {
      float4 pv = prow[k];
      float4 ev = ((const float4*)erow)[k];
      dot += pv.x * ev.x + pv.y * ev.y + pv.z * ev.z + pv.w * ev.w;
    }
    const float p2 = pn2p[((size_t)q * C_C + c) * M_D + m];
    const float dd = sqrtf(fmaxf(e2 + p2 - 2.f * dot, 0.f));
    mx = fmaxf(mx, dd);
  }
#pragma unroll
  for (int off = 16; off >= 1; off >>= 1) mx = fmaxf(mx, __shfl_xor(mx, off, 32));
  if (lane == 0) atomicAdd(&acc[1], (double)mx);
}

__global__ void finalize_kernel(const double* __restrict__ acc,
                                float* __restrict__ out) {
  if (blockIdx.x == 0 && threadIdx.x == 0) {
    out[0] = (float)(-acc[0] / ((double)N_S * (double)(C_C - 1) * 2.0));
    out[1] = (float)( acc[1] / ((double)N_S * (double)M_D * 2.0));
  }
}

extern "C" void kernel_launch(void* const* d_in, const int* in_sizes, int n_in,
                              void* d_out, int out_size, void* d_ws, size_t ws_size,
                              hipStream_t stream) {
  (void)in_sizes; (void)n_in; (void)out_size; (void)ws_size;
  const float* emb0 = (const float*)d_in[0];
  const float* emb1 = (const float*)d_in[1];
  const float* piv0 = (const float*)d_in[2];
  const float* piv1 = (const float*)d_in[3];
  const int*   ally = (const int*)d_in[4];
  const int*   alld = (const int*)d_in[5];
  float* out = (float*)d_out;

  // Workspace layout (bytes):
  //   [0,16)       acc: 2 double accumulators (rep_sum, att_sum)
  //   [64,68)      meta: tile count
  //   [128,1664)   perm: int[384]
  //   [2048,2360)  desc: int[26*3]
  //   [4096,7168)  en2: 2*N floats
  //   [8192,...)   pn2: 2*Q*C*M floats (~2.0 MB)
  char*   ws   = (char*)d_ws;
  double* acc  = (double*)ws;
  int*    meta = (int*)(ws + 64);
  int*    perm = (int*)(ws + 128);
  int*    desc = (int*)(ws + 2048);
  float*  en2  = (float*)(ws + 4096);
  float*  pn2  = (float*)(ws + 8192);

  hipMemsetAsync(acc, 0, 2 * sizeof(double), stream);

  partition_kernel<<<1, 32, 0, stream>>>(alld, perm, desc, meta);

  {  // emb norms: 2*N waves, 8 waves per block
    int waves = 2 * N_S;
    rowsq_kernel<<<(waves + 7) / 8, 256, 0, stream>>>(emb0, emb1, en2, N_S);
  }
  {  // pivot norms: 2*Q*C*M waves
    int waves = 2 * QCM;
    rowsq_kernel<<<(waves + 7) / 8, 256, 0, stream>>>(piv0, piv1, pn2, QCM);
  }

  dim3 gridMain(MAXT, C_C, 2);
  dist_wmma_kernel<<<gridMain, 32, 0, stream>>>(emb0, emb1, piv0, piv1,
                                                en2, pn2, ally,
                                                perm, desc, meta, acc);

  dim3 gridAtt(N_S, M_D, 2);
  att_kernel<<<gridAtt, 32, 0, stream>>>(emb0, emb1, piv0, piv1,
                                         en2, pn2, ally, acc);

  finalize_kernel<<<1, 32, 0, stream>>>(acc, out);
}